// SearchNeighbors_22952305230180
// MI455X (gfx1250) — compile-verified
//
#include <hip/hip_runtime.h>
#include <hip/hip_bf16.h>

// Problem constants (from reference)
#define B_ 128
#define N_ 512
#define D_ 256
#define S_ 8
#define NCHUNK 4
#define CH_ (N_ / NCHUNK)   // 128 rows per chunk

typedef __attribute__((ext_vector_type(2))) float v2f;
typedef __attribute__((ext_vector_type(8))) float v8f;

// ---------------------------------------------------------------------------
// K1: key_attn[b,n] = sum_s att_stack[b,n,s]*stack_ptr[b,s]
//     new_att_stack[b,n,s] = att_stack[b,n,s]*(1-stack_ptr[b,s])
//     new_stack_ptr (first 1024 threads)
// ---------------------------------------------------------------------------
__global__ void k1_stack(const float* __restrict__ att_stack,
                         const float* __restrict__ stack_ptr,
                         float* __restrict__ out_att,
                         float* __restrict__ out_ptr,
                         float* __restrict__ key_attn) {
    int gid = blockIdx.x * blockDim.x + threadIdx.x;   // [0, B*N)
    if (gid < B_ * N_) {
        int b = gid >> 9;                              // n = gid & 511
        const float4* av = (const float4*)(att_stack + (size_t)gid * S_);
        const float4* pv = (const float4*)(stack_ptr + (size_t)b * S_);
        float4 a0 = av[0], a1 = av[1];
        float4 p0 = pv[0], p1 = pv[1];
        float ka = a0.x*p0.x + a0.y*p0.y + a0.z*p0.z + a0.w*p0.w
                 + a1.x*p1.x + a1.y*p1.y + a1.z*p1.z + a1.w*p1.w;
        key_attn[gid] = ka;
        float4 o0, o1;
        o0.x = a0.x*(1.f-p0.x); o0.y = a0.y*(1.f-p0.y);
        o0.z = a0.z*(1.f-p0.z); o0.w = a0.w*(1.f-p0.w);
        o1.x = a1.x*(1.f-p1.x); o1.y = a1.y*(1.f-p1.y);
        o1.z = a1.z*(1.f-p1.z); o1.w = a1.w*(1.f-p1.w);
        float4* ov = (float4*)(out_att + (size_t)gid * S_);
        ov[0] = o0; ov[1] = o1;
    }
    // new_stack_ptr: new[0]=ptr[0]+ptr[1]; new[s]=ptr[s+1] (s<7); new[7]=0
    if (gid < B_ * S_) {
        int b = gid >> 3, s = gid & 7;
        float v = (s < S_ - 1) ? stack_ptr[b * S_ + s + 1] : 0.f;
        if (s == 0) v += stack_ptr[b * S_];
        out_ptr[gid] = v;
    }
}

// ---------------------------------------------------------------------------
// K2e: partial edge_matrix over an N-chunk (b128 streaming, the hot loop)
//   edge_part[c][b][m] = sum_{n in chunk c} edge[b,n,m]*ka[b,n]
// grid (B, NCHUNK), block 128; thread owns 4 consecutive columns (float4).
// ---------------------------------------------------------------------------
__global__ __launch_bounds__(128)
void k2_edge(const float* __restrict__ edge,
             const float* __restrict__ key_attn,
             float* __restrict__ edge_part) {
    __shared__ float s_ka[CH_];
    int b = blockIdx.x, c = blockIdx.y;
    int t = threadIdx.x;                     // 0..127
    int n0 = c * CH_;
    s_ka[t] = key_attn[(size_t)b * N_ + n0 + t];
    __syncthreads();

    const float4* eb = (const float4*)(edge + (size_t)b * N_ * N_
                                            + (size_t)n0 * N_) + t;
    float4 acc = {0.f, 0.f, 0.f, 0.f};
    #pragma unroll 4
    for (int n = 0; n < CH_; ++n) {
        float ka = s_ka[n];
        float4 v = eb[(size_t)n * (N_ / 4)];     // 128 lanes x 16B = full 2KB row
        acc.x += v.x * ka; acc.y += v.y * ka;
        acc.z += v.z * ka; acc.w += v.w * ka;
    }
    ((float4*)(edge_part + ((size_t)c * B_ + b) * N_))[t] = acc;
}

// ---------------------------------------------------------------------------
// K2f: partial key_feat over an N-chunk
//   kf_part[c][b][d] = sum_{n in chunk c} node_feat[b,n,d]*ka[b,n]
// grid (B, NCHUNK), block 128; thread owns 2 consecutive columns (float2).
// ---------------------------------------------------------------------------
__global__ __launch_bounds__(128)
void k2_feat(const float* __restrict__ node_feat,
             const float* __restrict__ key_attn,
             float* __restrict__ kf_part) {
    __shared__ float s_ka[CH_];
    int b = blockIdx.x, c = blockIdx.y;
    int t = threadIdx.x;                     // 0..127
    int n0 = c * CH_;
    s_ka[t] = key_attn[(size_t)b * N_ + n0 + t];
    __syncthreads();

    const float2* nb = (const float2*)(node_feat + (size_t)b * N_ * D_
                                                 + (size_t)n0 * D_) + t;
    float2 acc = {0.f, 0.f};
    #pragma unroll 4
    for (int n = 0; n < CH_; ++n) {
        float ka = s_ka[n];
        float2 v = nb[(size_t)n * (D_ / 2)];     // 128 lanes x 8B = full 1KB row
        acc.x += v.x * ka; acc.y += v.y * ka;
    }
    ((float2*)(kf_part + ((size_t)c * B_ + b) * D_))[t] = acc;
}

// ---------------------------------------------------------------------------
// K2c: deterministic combine of the NCHUNK partials
// ---------------------------------------------------------------------------
__global__ void k2_combine(const float* __restrict__ edge_part,
                           const float* __restrict__ kf_part,
                           float* __restrict__ edge_m,
                           float* __restrict__ key_feat) {
    int gid = blockIdx.x * blockDim.x + threadIdx.x;
    const int TE = B_ * N_;                  // 65536
    const int TF = B_ * D_;                  // 32768
    if (gid < TE) {
        float s = 0.f;
        #pragma unroll
        for (int c = 0; c < NCHUNK; ++c) s += edge_part[(size_t)c * TE + gid];
        edge_m[gid] = s;
    } else if (gid < TE + TF) {
        int i = gid - TE;
        float s = 0.f;
        #pragma unroll
        for (int c = 0; c < NCHUNK; ++c) s += kf_part[(size_t)c * TF + i];
        key_feat[i] = s;
    }
}

// ---------------------------------------------------------------------------
// K3: q[128,256] = key_feat[128,256] @ W^T + bias   via V_WMMA_F32_16X16X4_F32
// One wave per 16x16 output tile. grid = (M/16, Nout/16), block = 32 (wave32).
// A 16x4 layout : lanes 0-15 = rows M; lane-half selects K pair {0,1}/{2,3}
// B 4x16 layout : lanes 0-15 = cols N; lane-half selects K pair (mirror of A)
// C/D 16x16     : vgpr r -> M = r + 8*half, N = lane&15
// ---------------------------------------------------------------------------
__global__ __launch_bounds__(32)
void k3_wmma_gemm(const float* __restrict__ key_feat,   // [128,256] row-major
                  const float* __restrict__ W,          // [256,256] row-major [out,in]
                  const float* __restrict__ bias,       // [256]
                  float* __restrict__ q) {              // [128,256]
    int lane = threadIdx.x;
    int half = lane >> 4;          // 0: K={0,1}; 1: K={2,3}
    int lid  = lane & 15;
    int m0 = blockIdx.x * 16;      // batch-row tile
    int n0 = blockIdx.y * 16;      // output-dim tile

    const float* arow = key_feat + (size_t)(m0 + lid) * D_;  // A[m, :]
    const float* brow = W        + (size_t)(n0 + lid) * D_;  // B[k,n]=W[n,k]

    v8f acc = {};
    #pragma unroll 8
    for (int k = 0; k < D_; k += 4) {
        int ka = k + 2 * half;
        v2f a  = { arow[ka], arow[ka + 1] };
        v2f bb = { brow[ka], brow[ka + 1] };
        acc = __builtin_amdgcn_wmma_f32_16x16x4_f32(
            /*neg_a=*/false, a, /*neg_b=*/false, bb,
            /*c_mod=*/(short)0, acc, /*reuse_a=*/false, /*reuse_b=*/false);
    }

    float bn = bias[n0 + lid];
    #pragma unroll
    for (int r = 0; r < 8; ++r) {
        int m = m0 + r + 8 * half;
        q[(size_t)m * D_ + n0 + lid] = acc[r] + bn;
    }
}

// ---------------------------------------------------------------------------
// K4: new_value_mem[b,n] = value_mem[b,n] + edge_m[b,n]*dot(node_feat[b,n,:], q[b,:])
// One wave per row; lane covers 8 consecutive d via two float4 (b128) loads.
// ---------------------------------------------------------------------------
__global__ void k4_logits(const float* __restrict__ node_feat,
                          const float* __restrict__ qv,
                          const float* __restrict__ edge_m,
                          const float* __restrict__ value_mem,
                          float* __restrict__ out_vm) {
    int wave = (blockIdx.x * blockDim.x + threadIdx.x) >> 5;   // row id [0, B*N)
    int lane = threadIdx.x & 31;
    if (wave >= B_ * N_) return;
    int b = wave >> 9;

    const float4* nf = (const float4*)(node_feat + (size_t)wave * D_ + lane * 8);
    const float4* qp = (const float4*)(qv        + (size_t)b    * D_ + lane * 8);
    float4 n0 = nf[0], n1 = nf[1];
    float4 q0 = qp[0], q1 = qp[1];
    float v = n0.x*q0.x + n0.y*q0.y + n0.z*q0.z + n0.w*q0.w
            + n1.x*q1.x + n1.y*q1.y + n1.z*q1.z + n1.w*q1.w;
    #pragma unroll
    for (int off = 16; off > 0; off >>= 1)
        v += __shfl_xor(v, off, 32);
    if (lane == 0)
        out_vm[wave] = value_mem[wave] + edge_m[wave] * v;
}

// ---------------------------------------------------------------------------
extern "C" void kernel_launch(void* const* d_in, const int* in_sizes, int n_in,
                              void* d_out, int out_size, void* d_ws, size_t ws_size,
                              hipStream_t stream) {
    const float* node_feat = (const float*)d_in[0];
    // d_in[1] = query (unused by the reference)
    const float* edge      = (const float*)d_in[2];
    const float* att_stack = (const float*)d_in[3];
    const float* stack_ptr = (const float*)d_in[4];
    const float* value_mem = (const float*)d_in[5];
    const float* W         = (const float*)d_in[6];
    const float* bias      = (const float*)d_in[7];

    float* out_att = (float*)d_out;                    // B*N*S
    float* out_ptr = out_att + (size_t)B_ * N_ * S_;   // B*S
    float* out_vm  = out_ptr + (size_t)B_ * S_;        // B*N

    float* ws        = (float*)d_ws;
    float* key_attn  = ws;                                   // B*N      = 65536
    float* edge_m    = key_attn + (size_t)B_ * N_;           // B*N      = 65536
    float* key_feat  = edge_m   + (size_t)B_ * N_;           // B*D      = 32768
    float* qv        = key_feat + (size_t)B_ * D_;           // B*D      = 32768
    float* edge_part = qv       + (size_t)B_ * D_;           // 4*B*N    = 262144
    float* kf_part   = edge_part + (size_t)NCHUNK * B_ * N_; // 4*B*D    = 131072

    // K1: stack read/write + pointer shift
    k1_stack<<<(B_ * N_) / 256, 256, 0, stream>>>(att_stack, stack_ptr,
                                                  out_att, out_ptr, key_attn);
    // K2: chunked b128/b64 streaming reductions (dominant bandwidth), then combine
    k2_edge<<<dim3(B_, NCHUNK), 128, 0, stream>>>(edge, key_attn, edge_part);
    k2_feat<<<dim3(B_, NCHUNK), 128, 0, stream>>>(node_feat, key_attn, kf_part);
    k2_combine<<<(B_ * N_ + B_ * D_) / 256, 256, 0, stream>>>(edge_part, kf_part,
                                                              edge_m, key_feat);
    // K3: WMMA GEMM q = key_feat @ W^T + b
    k3_wmma_gemm<<<dim3(B_ / 16, D_ / 16), 32, 0, stream>>>(key_feat, W, bias, qv);
    // K4: logits dot + value_mem accumulate
    k4_logits<<<(B_ * N_) / 8, 256, 0, stream>>>(node_feat, qv, edge_m,
                                                 value_mem, out_vm);
}